// PagedMoePythonQwen35Experts_73684458930297
// MI455X (gfx1250) — compile-verified
//
#include <hip/hip_runtime.h>
#include <math.h>

#define T_TOK 128
#define K_TOP 8
#define H_DIM 2048
#define E_EXP 64
#define I_DIM 768

typedef float v2f __attribute__((ext_vector_type(2)));
typedef float v8f __attribute__((ext_vector_type(8)));

// act tile row stride in dwords: 768 + 4 pad so ds_load_b64 A-fetches in the
// down-projection hit distinct LDS banks (768 % 64 == 0 would be 16-way conflict)
#define ACT_STRIDE 772

// ---------------------------------------------------------------------------
// Kernel 1: route (token, weight) pairs into per-expert compact lists
// ---------------------------------------------------------------------------
__global__ void moe_route_kernel(const int* __restrict__ topk_idx,
                                 const float* __restrict__ topk_w,
                                 int* __restrict__ counts,
                                 int* __restrict__ toks,
                                 float* __restrict__ wts) {
    int tid = threadIdx.x;
    if (tid < E_EXP) counts[tid] = 0;
    __syncthreads();
    for (int i = tid; i < T_TOK * K_TOP; i += blockDim.x) {
        int e = topk_idx[i];
        int t = i / K_TOP;
        float w = topk_w[i];
        int pos = atomicAdd(&counts[e], 1);
        toks[e * T_TOK + pos] = t;
        wts[e * T_TOK + pos] = w;
    }
}

// ---------------------------------------------------------------------------
// Kernel 2: zero the output accumulator (scatter uses f32 atomics)
// ---------------------------------------------------------------------------
__global__ void moe_zero_out_kernel(float* __restrict__ out, int n) {
    int i = blockIdx.x * blockDim.x + threadIdx.x;
    if (i < n) out[i] = 0.f;
}

// ---------------------------------------------------------------------------
// Kernel 3: per-(expert, 16-token m-tile) fused gate/up/silu/down + scatter
// 256 threads = 8 wave32 waves. Stage 1: waves split I=768 into 8x96 cols.
// Stage 2: waves split H=2048 into 8x256 cols.
// ---------------------------------------------------------------------------
__global__ __launch_bounds__(256)
void moe_expert_kernel(const float* __restrict__ h,
                       const float* __restrict__ w_gate,
                       const float* __restrict__ w_up,
                       const float* __restrict__ w_down,
                       const int* __restrict__ counts,
                       const int* __restrict__ toks,
                       const float* __restrict__ wts,
                       float* __restrict__ out) {
    __shared__ float s_act[16 * ACT_STRIDE];
    __shared__ int   s_tok[16];
    __shared__ float s_wt[16];

    const int e  = blockIdx.x >> 3;
    const int mt = blockIdx.x & 7;
    const int n_e = counts[e];
    if (mt * 16 >= n_e) return;   // block-uniform exit, EXEC stays full below

    const int tid = threadIdx.x;
    if (tid < 16) {
        int idx = mt * 16 + tid;
        s_tok[tid] = (idx < n_e) ? toks[e * T_TOK + idx] : -1;
        s_wt[tid]  = (idx < n_e) ? wts[e * T_TOK + idx] : 0.f;
    }
    __syncthreads();

    const int wave = tid >> 5;
    const int lane = tid & 31;
    const int lm   = lane & 15;
    const int hi   = lane >> 4;   // 0: lanes 0-15, 1: lanes 16-31
    const int kk0  = hi * 2;      // A/B K sub-offset per ISA 16x4 layout

    int a_tok = s_tok[lm];
    const float* a_row = h + (size_t)(a_tok < 0 ? 0 : a_tok) * H_DIM + kk0;

    const float* Wg = w_gate + (size_t)e * H_DIM * I_DIM;
    const float* Wu = w_up   + (size_t)e * H_DIM * I_DIM;

    // ---------------- Stage 1: gate/up projection + SiLU, act -> LDS -------
    for (int nt = 0; nt < 6; ++nt) {
        const int ncol = wave * 96 + nt * 16 + lm;
        const float* wg = Wg + (size_t)kk0 * I_DIM + ncol;
        const float* wu = Wu + (size_t)kk0 * I_DIM + ncol;
        v8f accg = {};
        v8f accu = {};
#pragma unroll 4
        for (int k = 0; k < H_DIM; k += 4) {
            v2f a, bg, bu;
            a.x  = a_row[k];
            a.y  = a_row[k + 1];
            bg.x = wg[(size_t)k * I_DIM];
            bg.y = wg[(size_t)(k + 1) * I_DIM];
            bu.x = wu[(size_t)k * I_DIM];
            bu.y = wu[(size_t)(k + 1) * I_DIM];
            accg = __builtin_amdgcn_wmma_f32_16x16x4_f32(
                false, a, false, bg, (short)0, accg, false, false);
            accu = __builtin_amdgcn_wmma_f32_16x16x4_f32(
                false, a, false, bu, (short)0, accu, false, false);
        }
#pragma unroll
        for (int r = 0; r < 8; ++r) {
            float g = accg[r];
            float u = accu[r];
            // silu via v_exp_f32 + v_rcp_f32 (avoid IEEE div_scale/div_fixup chain)
            float sg = g * __builtin_amdgcn_rcpf(1.f + __expf(-g));
            int row  = r + hi * 8;               // C/D layout: vgpr r -> rows r, r+8
            s_act[row * ACT_STRIDE + wave * 96 + nt * 16 + lm] = sg * u;
        }
    }
    __syncthreads();

    // ---------------- Stage 2: down projection + weighted scatter ----------
    const float* Wd = w_down + (size_t)e * I_DIM * H_DIM;
    const float* arow = &s_act[lm * ACT_STRIDE + kk0];
    for (int nt = 0; nt < 16; ++nt) {
        const int ncol = wave * 256 + nt * 16 + lm;
        const float* wd = Wd + (size_t)kk0 * H_DIM + ncol;
        v8f acc = {};
#pragma unroll 4
        for (int k = 0; k < I_DIM; k += 4) {
            v2f a, b;
            a.x = arow[k];
            a.y = arow[k + 1];
            b.x = wd[(size_t)k * H_DIM];
            b.y = wd[(size_t)(k + 1) * H_DIM];
            acc = __builtin_amdgcn_wmma_f32_16x16x4_f32(
                false, a, false, b, (short)0, acc, false, false);
        }
#pragma unroll
        for (int r = 0; r < 8; ++r) {
            int row = r + hi * 8;
            int tok = s_tok[row];
            if (tok >= 0) {
                atomicAdd(&out[(size_t)tok * H_DIM + ncol], s_wt[row] * acc[r]);
            }
        }
    }
}

// ---------------------------------------------------------------------------
extern "C" void kernel_launch(void* const* d_in, const int* in_sizes, int n_in,
                              void* d_out, int out_size, void* d_ws, size_t ws_size,
                              hipStream_t stream) {
    const float* h      = (const float*)d_in[0];
    const int*   topk_i = (const int*)d_in[1];
    const float* topk_w = (const float*)d_in[2];
    const float* w_gate = (const float*)d_in[3];
    const float* w_up   = (const float*)d_in[4];
    const float* w_down = (const float*)d_in[5];
    float* out = (float*)d_out;

    char* ws = (char*)d_ws;
    int*   counts = (int*)ws;                                   // E ints
    int*   toks   = (int*)(ws + 256);                           // E*T ints
    float* wts    = (float*)(ws + 256 + E_EXP * T_TOK * 4);     // E*T floats

    moe_route_kernel<<<1, 256, 0, stream>>>(topk_i, topk_w, counts, toks, wts);

    int n_out = T_TOK * H_DIM;
    moe_zero_out_kernel<<<(n_out + 255) / 256, 256, 0, stream>>>(out, n_out);

    moe_expert_kernel<<<E_EXP * 8, 256, 0, stream>>>(
        h, w_gate, w_up, w_down, counts, toks, wts, out);
}